// DeformableTransformer_412316860435
// MI455X (gfx1250) — compile-verified
//
#include <hip/hip_runtime.h>
#include <hip/hip_bf16.h>

// ---------------------------------------------------------------------------
// Deformable attention (MSDeformAttn) for MI455X / gfx1250, wave32 + WMMA.
//   GEMMs       : V_WMMA_F32_16X16X4_F32 (fp32 end-to-end), K=256 looped.
//                 B-panel staged LDS via GLOBAL_LOAD_ASYNC_TO_LDS_B128
//                 (ASYNCcnt path), 2 column tiles per wave per A fragment.
//   Softmax     : 1 thread / 16-logit row
//   Grid-sample : 1 wave32 / (query, head); lane == channel (DH == 32)
// ---------------------------------------------------------------------------

typedef float v2f __attribute__((ext_vector_type(2)));
typedef float v8f __attribute__((ext_vector_type(8)));

#define LEN_IN   19947
#define DMODEL   256
#define NHEADS   8

#if defined(__has_builtin)
#  if __has_builtin(__builtin_amdgcn_s_wait_asynccnt)
#    define WAIT_ASYNC0() __builtin_amdgcn_s_wait_asynccnt(0)
#  endif
#endif
#ifndef WAIT_ASYNC0
#  define WAIT_ASYNC0() asm volatile("s_wait_asynccnt 0" ::: "memory")
#endif

// ---------------------------------------------------------------------------
// C[M x N] = A[M x 256] @ W[256 x N] + bias[N]; optional row mask zeroes rows.
// Block: 128 threads = 4 waves; block tile 64(M) x 32(N); wave tile 16x32
// (two 16x16 WMMA column tiles sharing one A fragment).
// B-panel (256x32 fp32 = 32 KB) staged in LDS with async-to-LDS loads.
// ---------------------------------------------------------------------------
__global__ __launch_bounds__(128)
void wmma_gemm_k256(const float* __restrict__ A,
                    const float* __restrict__ W,
                    const float* __restrict__ bias,
                    const unsigned char* __restrict__ row_mask,
                    float* __restrict__ C,
                    int Mrows, int Ncols)
{
    __shared__ float Ws[256 * 32];   // B panel, row-major [k][c], 32 KB

    const int tid  = threadIdx.x;
    const int wave = tid >> 5;
    const int lane = tid & 31;
    const int half = lane >> 4;   // 0: K+{0,1} / rows 0-7 ; 1: K+{2,3} / rows 8-15
    const int r    = lane & 15;

    const int col0 = blockIdx.y * 32;
    const int row0 = blockIdx.x * 64 + wave * 16;

    // ---- Async global -> LDS staging of the 256x32 W panel ----------------
    // GLOBAL_LOAD_ASYNC_TO_LDS_B128, GV form: vdst = 32-bit LDS byte address,
    // vaddr = 64-bit global address. Tracked with ASYNCcnt (no VGPR data).
    {
        const unsigned lds_base = (unsigned)(uintptr_t)(&Ws[0]);
        for (int i = tid; i < 256 * 8; i += 128) {       // 2048 b128 chunks
            const int k  = i >> 3;
            const int c4 = (i & 7) * 4;
            const float*   g = W + (size_t)k * Ncols + col0 + c4;
            const unsigned l = lds_base + (unsigned)((k * 32 + c4) * 4);
            asm volatile("global_load_async_to_lds_b128 %0, %1, off"
                         :: "v"(l), "v"(g)
                         : "memory");
        }
        WAIT_ASYNC0();      // s_wait_asynccnt 0
    }
    __syncthreads();

    // A fragment source row for this lane (clamped; OOB rows never stored).
    int arow = row0 + r;
    if (arow >= Mrows) arow = Mrows - 1;
    const float* __restrict__ Aptr = A + (size_t)arow * 256;

    v8f acc0 = {0.f, 0.f, 0.f, 0.f, 0.f, 0.f, 0.f, 0.f};
    v8f acc1 = {0.f, 0.f, 0.f, 0.f, 0.f, 0.f, 0.f, 0.f};

    for (int kk = 0; kk < 256; kk += 4) {
        // gfx1250 prefetch of upcoming A data (global_prefetch_b8), near scope.
        __builtin_prefetch(Aptr + kk + 64, 0, 3);

        const int ka = kk + half * 2;
        v2f a, b0, b1;
        // A 16x4 fragment: lanes 0-15 K=kk+{0,1}; lanes 16-31 K=kk+{2,3}.
        a.x = Aptr[ka];
        a.y = Aptr[ka + 1];
        // Two B 4x16 fragments (mirrored layout) from LDS: cols [0,16) & [16,32).
        b0.x = Ws[ka * 32 + r];
        b0.y = Ws[(ka + 1) * 32 + r];
        b1.x = Ws[ka * 32 + 16 + r];
        b1.y = Ws[(ka + 1) * 32 + 16 + r];

        acc0 = __builtin_amdgcn_wmma_f32_16x16x4_f32(
            false, a, false, b0, (short)0, acc0, false, false);
        acc1 = __builtin_amdgcn_wmma_f32_16x16x4_f32(
            false, a, false, b1, (short)0, acc1, false, false);
    }

    // D 16x16 layout: lane -> column, VGPR i -> row (half*8 + i).
    const float bc0 = bias ? bias[col0 + r]      : 0.f;
    const float bc1 = bias ? bias[col0 + 16 + r] : 0.f;
    for (int i = 0; i < 8; ++i) {
        const int row = row0 + half * 8 + i;
        if (row < Mrows) {
            float v0 = acc0[i] + bc0;
            float v1 = acc1[i] + bc1;
            if (row_mask && row_mask[row]) { v0 = 0.f; v1 = 0.f; }
            C[(size_t)row * Ncols + col0 + r]      = v0;
            C[(size_t)row * Ncols + col0 + 16 + r] = v1;
        }
    }
}

// ---------------------------------------------------------------------------
// Row softmax over 16 logits (one thread per (q, head) row).
// ---------------------------------------------------------------------------
__global__ __launch_bounds__(256)
void softmax16(float* __restrict__ attn, int rows)
{
    const int row = blockIdx.x * blockDim.x + threadIdx.x;
    if (row >= rows) return;
    float* p = attn + (size_t)row * 16;

    float v[16];
    float mx = -1e30f;
#pragma unroll
    for (int i = 0; i < 16; ++i) { v[i] = p[i]; mx = fmaxf(mx, v[i]); }
    float s = 0.f;
#pragma unroll
    for (int i = 0; i < 16; ++i) { v[i] = __expf(v[i] - mx); s += v[i]; }
    const float inv = 1.f / s;
#pragma unroll
    for (int i = 0; i < 16; ++i) p[i] = v[i] * inv;
}

// ---------------------------------------------------------------------------
// Bilinear grid-sample + attention-weighted accumulation.
// One wave32 per (q, head); lane == channel. Each corner fetch of a head's
// 32-channel slice is one coalesced 128B line (wave32 fits DH=32 exactly).
// ---------------------------------------------------------------------------
__global__ __launch_bounds__(256)
void deform_sample(const float* __restrict__ ref,    // (LEN_IN, 4, 2)
                   const float* __restrict__ value,  // (LEN_IN, 256)
                   const float* __restrict__ off,    // (LEN_IN, 256) = (m,l,p,2)
                   const float* __restrict__ aw,     // (LEN_IN, 128) = (m,l,p)
                   float* __restrict__ tmp)          // (LEN_IN, 256)
{
    const int wave = threadIdx.x >> 5;
    const int lane = threadIdx.x & 31;
    const int gid  = blockIdx.x * 8 + wave;           // q*8 + m
    if (gid >= LEN_IN * NHEADS) return;
    const int q = gid >> 3;
    const int m = gid & 7;

    const int Hs[4]     = {100, 50, 25, 13};
    const int Wd[4]     = {150, 75, 38, 19};
    const int starts[4] = {0, 15000, 18750, 19700};

    const float* __restrict__ offq  = off + (size_t)q * DMODEL + m * 32;
    const float* __restrict__ awq   = aw  + (size_t)q * 128   + m * 16;
    const float* __restrict__ refq  = ref + (size_t)q * 8;
    const float* __restrict__ vbase = value + (size_t)m * 32 + lane;

    float acc = 0.f;

#pragma unroll
    for (int l = 0; l < 4; ++l) {
        const int   H    = Hs[l];
        const int   Wl   = Wd[l];
        const int   st   = starts[l];
        const float rx   = refq[l * 2 + 0];
        const float ry   = refq[l * 2 + 1];
        const float invW = 1.f / (float)Wl;
        const float invH = 1.f / (float)H;

#pragma unroll
        for (int p = 0; p < 4; ++p) {
            const float ox = offq[(l * 4 + p) * 2 + 0];
            const float oy = offq[(l * 4 + p) * 2 + 1];
            float gx = 2.f * (rx + ox * invW) - 1.f;
            float gy = 2.f * (ry + oy * invH) - 1.f;
            gx = fminf(fmaxf(gx, -2.f), 2.f);
            gy = fminf(fmaxf(gy, -2.f), 2.f);

            const float x = (gx + 1.f) * (0.5f * (float)Wl) - 0.5f;
            const float y = (gy + 1.f) * (0.5f * (float)H)  - 0.5f;
            const float x0f = floorf(x);
            const float y0f = floorf(y);
            const int   x0  = (int)x0f;
            const int   y0  = (int)y0f;
            const float wx1 = x - x0f, wx0 = 1.f - wx1;
            const float wy1 = y - y0f, wy0 = 1.f - wy1;

            float s = 0.f;
#pragma unroll
            for (int c = 0; c < 4; ++c) {
                const int   xi = x0 + (c & 1);
                const int   yi = y0 + (c >> 1);
                const float w  = ((c & 1) ? wx1 : wx0) * ((c >> 1) ? wy1 : wy0);
                if (xi >= 0 && xi < Wl && yi >= 0 && yi < H) {
                    s += w * vbase[(size_t)(st + yi * Wl + xi) * DMODEL];
                }
            }
            acc += awq[l * 4 + p] * s;
        }
    }

    tmp[(size_t)q * DMODEL + m * 32 + lane] = acc;
}

// ---------------------------------------------------------------------------
extern "C" void kernel_launch(void* const* d_in, const int* in_sizes, int n_in,
                              void* d_out, int out_size, void* d_ws, size_t ws_size,
                              hipStream_t stream)
{
    const float*         query   = (const float*)d_in[0];
    const float*         refpts  = (const float*)d_in[1];
    const float*         in_flat = (const float*)d_in[2];
    const unsigned char* mask    = (const unsigned char*)d_in[3];
    const float*         W_off   = (const float*)d_in[4];
    const float*         b_off   = (const float*)d_in[5];
    const float*         W_attn  = (const float*)d_in[6];
    const float*         b_attn  = (const float*)d_in[7];
    const float*         W_val   = (const float*)d_in[8];
    const float*         b_val   = (const float*)d_in[9];
    const float*         W_out   = (const float*)d_in[10];
    const float*         b_out   = (const float*)d_in[11];
    float*               out     = (float*)d_out;

    // Workspace carve-up (floats): value | off | attn | tmp  (~71.5 MB total)
    float* ws    = (float*)d_ws;
    float* value = ws;
    float* off   = value + (size_t)LEN_IN * DMODEL;
    float* attn  = off   + (size_t)LEN_IN * DMODEL;
    float* tmp   = attn  + (size_t)LEN_IN * 128;

    const dim3 blk(128);
    const int  gx = (LEN_IN + 63) / 64;

    // 1) value = input_flatten @ W_val + b_val, masked rows -> 0
    wmma_gemm_k256<<<dim3(gx, 8), blk, 0, stream>>>(in_flat, W_val, b_val, mask, value, LEN_IN, 256);
    // 2) off = query @ W_off + b_off
    wmma_gemm_k256<<<dim3(gx, 8), blk, 0, stream>>>(query, W_off, b_off, nullptr, off, LEN_IN, 256);
    // 3) attn logits = query @ W_attn + b_attn
    wmma_gemm_k256<<<dim3(gx, 4), blk, 0, stream>>>(query, W_attn, b_attn, nullptr, attn, LEN_IN, 128);
    // 4) softmax over 16 per (q, head)
    softmax16<<<(LEN_IN * NHEADS + 255) / 256, 256, 0, stream>>>(attn, LEN_IN * NHEADS);
    // 5) bilinear sampling + weighted sum
    deform_sample<<<(LEN_IN * NHEADS + 7) / 8, 256, 0, stream>>>(refpts, value, off, attn, tmp);
    // 6) out = tmp @ W_out + b_out
    wmma_gemm_k256<<<dim3(gx, 8), blk, 0, stream>>>(tmp, W_out, b_out, nullptr, out, LEN_IN, 256);
}